// KBBlock_s_21397527068799
// MI455X (gfx1250) — compile-verified
//
#include <hip/hip_runtime.h>
#include <cstddef>

typedef __attribute__((ext_vector_type(2))) float v2f;
typedef __attribute__((ext_vector_type(8))) float v8f;

#define PIX 16384      // H*W = 128*128
#define HH 128
#define WW 128
#define BB 2
#define CC 64

__device__ __forceinline__ v8f wmma_f32(v2f a, v2f b, v8f c) {
  // V_WMMA_F32_16X16X4_F32: (neg_a, A, neg_b, B, c_mod, C, reuse_a, reuse_b)
  return __builtin_amdgcn_wmma_f32_16x16x4_f32(false, a, false, b, (short)0, c,
                                               false, false);
}

// ---------------------------------------------------------------------------
// LayerNorm over channels, per pixel. grid = B*PIX/256
// ---------------------------------------------------------------------------
__global__ void __launch_bounds__(256)
ln_kernel(const float* __restrict__ in, const float* __restrict__ w,
          const float* __restrict__ bi, float* __restrict__ out) {
  int idx = blockIdx.x * 256 + threadIdx.x;     // b*PIX + p
  int b = idx >> 14;
  int p = idx & (PIX - 1);
  const float* base = in + ((size_t)b * CC) * PIX + p;
  float s = 0.f, s2 = 0.f;
  for (int c = 0; c < CC; ++c) {
    float v = base[(size_t)c * PIX];
    s += v; s2 += v * v;
  }
  float mu = s * (1.f / 64.f);
  float var = s2 * (1.f / 64.f) - mu * mu;
  float rs = rsqrtf(var + 1e-6f);
  float* ob = out + ((size_t)b * CC) * PIX + p;
  for (int c = 0; c < CC; ++c) {
    float v = base[(size_t)c * PIX];
    ob[(size_t)c * PIX] = w[c] * (v - mu) * rs + bi[c];
  }
}

// ---------------------------------------------------------------------------
// Per-(b,c) spatial mean. grid = B*C blocks of 256
// ---------------------------------------------------------------------------
__global__ void __launch_bounds__(256)
chmean_kernel(const float* __restrict__ x, float* __restrict__ xmean) {
  __shared__ float red[256];
  int bc = blockIdx.x;
  const float* base = x + (size_t)bc * PIX;
  float s = 0.f;
  for (int i = threadIdx.x; i < PIX; i += 256) s += base[i];
  red[threadIdx.x] = s;
  __syncthreads();
  for (int st = 128; st > 0; st >>= 1) {
    if ((int)threadIdx.x < st) red[threadIdx.x] += red[threadIdx.x + st];
    __syncthreads();
  }
  if (threadIdx.x == 0) xmean[bc] = red[0] * (1.f / (float)PIX);
}

// sca[b][co] = sca_b[co] + sum_ci sca_w[co][ci]*xmean[b][ci]   (1 block, 128 thr)
__global__ void sca_kernel(const float* __restrict__ w, const float* __restrict__ bi,
                           const float* __restrict__ xmean, float* __restrict__ sca) {
  int t = threadIdx.x;
  if (t >= BB * CC) return;
  int b = t >> 6, co = t & 63;
  float s = bi[co];
  for (int ci = 0; ci < CC; ++ci) s += w[co * 64 + ci] * xmean[b * 64 + ci];
  sca[t] = s;
}

// ---------------------------------------------------------------------------
// 1x1 conv as WMMA GEMM: OUT[Co,P] = W[Co,K] * IN[K,P] (+ fused epilogues)
// Wave owns MT 16-row tiles x one 16-pixel column. Block = 8 waves = 128 px.
// EPI 0: out = acc+bias ; EPI 1: out = res + (acc+bias)*scale
// EPI 2 (MT=8): SimpleGate -> out[c] = (acc_m+b[c])*(acc_{m+4}+b[c+64]), c<64
// ---------------------------------------------------------------------------
template <int MT, int EPI>
__global__ void __launch_bounds__(256)
conv1x1_wmma_kernel(const float* __restrict__ in, const float* __restrict__ w,
                    const float* __restrict__ bias, float* __restrict__ out,
                    int K, const float* __restrict__ res,
                    const float* __restrict__ scale) {
  const int lane = threadIdx.x & 31;
  const int wave = threadIdx.x >> 5;
  const int half = lane >> 4;
  const int lrow = lane & 15;
  const int b = blockIdx.y;
  const int col0 = blockIdx.x * 128 + wave * 16;
  const float* inb = in + (size_t)b * K * PIX;

  v8f acc[MT] = {};
  for (int kk = 0; kk < K; kk += 4) {
    v2f bf;
    const float* bp = inb + (size_t)(kk + 2 * half) * PIX + col0 + lrow;
    bf.x = bp[0];
    bf.y = bp[PIX];
#pragma unroll
    for (int m = 0; m < MT; ++m) {
      const float* ap = w + (size_t)(m * 16 + lrow) * K + kk + 2 * half;
      v2f af;
      af.x = ap[0];
      af.y = ap[1];
      acc[m] = wmma_f32(af, bf, acc[m]);
    }
  }

  const int p = col0 + lrow;
  if (EPI == 2) {  // gate: 128 -> 64 channels
    float* ob = out + ((size_t)b * 64) * PIX;
#pragma unroll
    for (int m = 0; m < 4; ++m)
#pragma unroll
      for (int r = 0; r < 8; ++r) {
        int c = m * 16 + r + 8 * half;
        float v1 = acc[m][r] + bias[c];
        float v2 = acc[m + 4][r] + bias[c + 64];
        ob[(size_t)c * PIX + p] = v1 * v2;
      }
  } else {
    const int Mout = MT * 16;
    float* ob = out + (size_t)b * Mout * PIX;
#pragma unroll
    for (int m = 0; m < MT; ++m)
#pragma unroll
      for (int r = 0; r < 8; ++r) {
        int c = m * 16 + r + 8 * half;
        float v = acc[m][r] + bias[c];
        if (EPI == 1) v = res[((size_t)b * Mout + c) * PIX + p] + v * scale[c];
        ob[(size_t)c * PIX + p] = v;
      }
  }
}

// ---------------------------------------------------------------------------
// 5x5 conv, groups=16 (Cin/g = 4). grid = B*64*PIX/256
// ---------------------------------------------------------------------------
__global__ void __launch_bounds__(256)
conv5x5_g16_kernel(const float* __restrict__ in, const float* __restrict__ w,
                   const float* __restrict__ bias, float* __restrict__ out) {
  int idx = blockIdx.x * 256 + threadIdx.x;
  int p = idx & (PIX - 1);
  int co = (idx >> 14) & 63;
  int b = idx >> 20;
  int py = p >> 7, px = p & 127;
  int g = co >> 2;
  const float* inb = in + ((size_t)b * CC + g * 4) * PIX;
  const float* wp = w + co * 100;
  float s = bias[co];
  for (int ci = 0; ci < 4; ++ci)
    for (int ky = 0; ky < 5; ++ky) {
      int yy = py + ky - 2;
      if (yy < 0 || yy > 127) continue;
      for (int kx = 0; kx < 5; ++kx) {
        int xx = px + kx - 2;
        if (xx < 0 || xx > 127) continue;
        s += inb[(size_t)ci * PIX + yy * WW + xx] * wp[(ci * 5 + ky) * 5 + kx];
      }
    }
  out[idx] = s;
}

// ---------------------------------------------------------------------------
// 3x3 conv groups=32 (Cin/g=2) fused with SimpleGate. grid = B*16*PIX/256
// ---------------------------------------------------------------------------
__global__ void __launch_bounds__(256)
conv3x3g32_gate_kernel(const float* __restrict__ in, const float* __restrict__ w,
                       const float* __restrict__ bias, float* __restrict__ out) {
  int idx = blockIdx.x * 256 + threadIdx.x;
  int p = idx & (PIX - 1);
  int j = (idx >> 14) & 15;
  int b = idx >> 18;
  int py = p >> 7, px = p & 127;
  float r[2];
#pragma unroll
  for (int h = 0; h < 2; ++h) {
    int m = j + h * 16;
    const float* inb = in + ((size_t)b * CC + m * 2) * PIX;
    const float* wp = w + m * 18;
    float s = bias[m];
    for (int ci = 0; ci < 2; ++ci)
      for (int ky = 0; ky < 3; ++ky) {
        int yy = py + ky - 1;
        if (yy < 0 || yy > 127) continue;
        for (int kx = 0; kx < 3; ++kx) {
          int xx = px + kx - 1;
          if (xx < 0 || xx > 127) continue;
          s += inb[(size_t)ci * PIX + yy * WW + xx] * wp[(ci * 3 + ky) * 3 + kx];
        }
      }
    r[h] = s;
  }
  out[((size_t)b * 16 + j) * PIX + p] = r[0] * r[1];
}

// ---------------------------------------------------------------------------
// depthwise 3x3. grid = B*64*PIX/256
// ---------------------------------------------------------------------------
__global__ void __launch_bounds__(256)
dw3x3_kernel(const float* __restrict__ in, const float* __restrict__ w,
             const float* __restrict__ bias, float* __restrict__ out) {
  int idx = blockIdx.x * 256 + threadIdx.x;
  int p = idx & (PIX - 1);
  int c = (idx >> 14) & 63;
  int b = idx >> 20;
  int py = p >> 7, px = p & 127;
  const float* inb = in + ((size_t)b * CC + c) * PIX;
  const float* wp = w + c * 9;
  float s = bias[c];
  for (int ky = 0; ky < 3; ++ky) {
    int yy = py + ky - 1;
    if (yy < 0 || yy > 127) continue;
    for (int kx = 0; kx < 3; ++kx) {
      int xx = px + kx - 1;
      if (xx < 0 || xx > 127) continue;
      s += inb[yy * WW + xx] * wp[ky * 3 + kx];
    }
  }
  out[idx] = s;
}

// ---------------------------------------------------------------------------
// KBA: per 16-pixel tile, per group g:
//   Y[(n*4+i), p] = sum_j W4[n,g,i,j] * uf[p,g,j]   (WMMA, M=128,K=36,N=16)
//   out[p,g,i]    = sum_n att[p,n] * Y[...] + att·kba_b
// Fused epilogue: xk = (kba*ga1 + uf) * x1 * sca.  att built from
// s2c*attgamma + c211 while staging into LDS.
// Block = 256 thr (8 waves); wave handles groups {w, w+8}. grid = B*H*(W/16)
// ---------------------------------------------------------------------------
__global__ void __launch_bounds__(256)
kba_kernel(const float* __restrict__ uf, const float* __restrict__ s2c,
           const float* __restrict__ c211o, const float* __restrict__ kba_w,
           const float* __restrict__ kba_b, const float* __restrict__ attgamma,
           const float* __restrict__ ga1, const float* __restrict__ x1,
           const float* __restrict__ sca, float* __restrict__ xk) {
  __shared__ float s_uf[64][3][18];   // halo patch, zero-padded
  __shared__ float s_att[32][16];     // att[n][px]

  int tile = blockIdx.x;
  int tx = tile & 7;
  int ty = (tile >> 3) & 127;
  int b = tile >> 10;
  int x0 = tx * 16;

  const float* ufb = uf + (size_t)b * CC * PIX;
  for (int i = threadIdx.x; i < 64 * 3 * 18; i += 256) {
    int ch = i / 54;
    int rr = i % 54;
    int ry = rr / 18, rx = rr % 18;
    int yy = ty + ry - 1, xx = x0 + rx - 1;
    float v = 0.f;
    if (yy >= 0 && yy < HH && xx >= 0 && xx < WW) v = ufb[(size_t)ch * PIX + yy * WW + xx];
    s_uf[ch][ry][rx] = v;
  }
  const float* s2b = s2c + (size_t)b * 32 * PIX;
  const float* c2b = c211o + (size_t)b * 32 * PIX;
  for (int i = threadIdx.x; i < 32 * 16; i += 256) {
    int n = i >> 4, px = i & 15;
    int pix = ty * WW + x0 + px;
    s_att[n][px] = s2b[(size_t)n * PIX + pix] * attgamma[n] + c2b[(size_t)n * PIX + pix];
  }
  __syncthreads();

  const int lane = threadIdx.x & 31;
  const int wave = threadIdx.x >> 5;
  const int half = lane >> 4;
  const int lrow = lane & 15;
  const int pix = ty * WW + x0 + lrow;

  for (int gi = 0; gi < 2; ++gi) {
    int g = wave + gi * 8;
    v8f acc[8] = {};
#pragma unroll
    for (int kk = 0; kk < 36; kk += 4) {
      v2f bf;
      {
        int j0 = kk + 2 * half;
        int cg0 = j0 / 9, r0 = j0 % 9;
        bf.x = s_uf[g * 4 + cg0][r0 / 3][lrow + r0 % 3];
        int j1 = j0 + 1;
        int cg1 = j1 / 9, r1 = j1 % 9;
        bf.y = s_uf[g * 4 + cg1][r1 / 3][lrow + r1 % 3];
      }
#pragma unroll
      for (int m = 0; m < 8; ++m) {
        int arow = m * 16 + lrow;          // M = n*4 + i
        int n = arow >> 2, ii = arow & 3;
        const float* ap = kba_w + (size_t)n * 2304 + g * 144 + ii * 36 + kk + 2 * half;
        v2f af;
        af.x = ap[0];
        af.y = ap[1];
        acc[m] = wmma_f32(af, bf, acc[m]);
      }
    }
    // contract over n with attention weights (lane halves hold disjoint n sets)
    float part[4] = {0.f, 0.f, 0.f, 0.f};
#pragma unroll
    for (int m = 0; m < 8; ++m)
#pragma unroll
      for (int r = 0; r < 8; ++r) {
        int Mv = m * 16 + r + 8 * half;
        int n = Mv >> 2, ii = Mv & 3;
        part[ii] += s_att[n][lrow] * acc[m][r];
      }
#pragma unroll
    for (int ii = 0; ii < 4; ++ii) {
      float tot = part[ii] + __shfl_xor(part[ii], 16, 32);
      if (half == 0) {
        int c = g * 4 + ii;
        float bsum = 0.f;
        for (int n = 0; n < 32; ++n) bsum += s_att[n][lrow] * kba_b[n * 64 + c];
        float kout = tot + bsum;
        float ufc = s_uf[c][1][lrow + 1];
        float v = (kout * ga1[c] + ufc) * x1[((size_t)b * CC + c) * PIX + pix] *
                  sca[b * 64 + c];
        xk[((size_t)b * CC + c) * PIX + pix] = v;
      }
    }
  }
}

// ---------------------------------------------------------------------------
extern "C" void kernel_launch(void* const* d_in, const int* in_sizes, int n_in,
                              void* d_out, int out_size, void* d_ws, size_t ws_size,
                              hipStream_t stream) {
  (void)in_sizes; (void)n_in; (void)out_size; (void)ws_size;
  const float* inp      = (const float*)d_in[0];
  const float* norm1_w  = (const float*)d_in[1];
  const float* norm1_b  = (const float*)d_in[2];
  const float* norm2_w  = (const float*)d_in[3];
  const float* norm2_b  = (const float*)d_in[4];
  const float* sca_w    = (const float*)d_in[5];
  const float* sca_b    = (const float*)d_in[6];
  const float* c11a_w   = (const float*)d_in[7];
  const float* c11a_b   = (const float*)d_in[8];
  const float* c11b_w   = (const float*)d_in[9];
  const float* c11b_b   = (const float*)d_in[10];
  const float* c1_w     = (const float*)d_in[11];
  const float* c1_b     = (const float*)d_in[12];
  const float* c21_w    = (const float*)d_in[13];
  const float* c21_b    = (const float*)d_in[14];
  const float* c2a_w    = (const float*)d_in[15];
  const float* c2a_b    = (const float*)d_in[16];
  const float* c2c_w    = (const float*)d_in[17];
  const float* c2c_b    = (const float*)d_in[18];
  const float* c211_w   = (const float*)d_in[19];
  const float* c211_b   = (const float*)d_in[20];
  const float* c3_w     = (const float*)d_in[21];
  const float* c3_b     = (const float*)d_in[22];
  const float* c4_w     = (const float*)d_in[23];
  const float* c4_b     = (const float*)d_in[24];
  const float* c5_w     = (const float*)d_in[25];
  const float* c5_b     = (const float*)d_in[26];
  const float* kba_w    = (const float*)d_in[27];
  const float* kba_b    = (const float*)d_in[28];
  const float* ga1      = (const float*)d_in[29];
  const float* attgamma = (const float*)d_in[30];
  const float* beta     = (const float*)d_in[31];
  const float* gamma    = (const float*)d_in[32];
  float* out = (float*)d_out;

  float* ws = (float*)d_ws;
  const size_t NBUF = (size_t)BB * CC * PIX;  // 2,097,152 floats / 8 MiB
  float* b_x    = ws;                          // LN1 out, reused as xk
  float* b_t1   = ws + 1 * NBUF;               // c11a out, reused as z (LN2 out)
  float* b_x1   = ws + 2 * NBUF;
  float* b_uf0  = ws + 3 * NBUF;               // reused as gated z
  float* b_uf   = ws + 4 * NBUF;
  float* b_y    = ws + 5 * NBUF;
  float* b_t    = ws + 6 * NBUF;               // [B,16,PIX]
  float* b_s2c  = b_t + (size_t)BB * 16 * PIX; // [B,32,PIX]
  float* b_c211 = b_s2c + (size_t)BB * 32 * PIX;
  float* b_xm   = b_c211 + (size_t)BB * 32 * PIX;  // [B*64]
  float* b_sca  = b_xm + BB * CC;
  float* b_xk = b_x;
  float* b_z = b_t1;
  float* b_zg = b_uf0;

  dim3 gemmGrid(PIX / 128, BB);

  // 1. x = LN1(inp)
  ln_kernel<<<BB * PIX / 256, 256, 0, stream>>>(inp, norm1_w, norm1_b, b_x);
  // 2-3. sca = 1x1(mean(x))
  chmean_kernel<<<BB * CC, 256, 0, stream>>>(b_x, b_xm);
  sca_kernel<<<1, 128, 0, stream>>>(sca_w, sca_b, b_xm, b_sca);
  // 4. t1 = c11a(x)
  conv1x1_wmma_kernel<4, 0><<<gemmGrid, 256, 0, stream>>>(b_x, c11a_w, c11a_b,
                                                          b_t1, 64, nullptr, nullptr);
  // 5. x1 = c11b(t1) (5x5, g=16)
  conv5x5_g16_kernel<<<BB * CC * PIX / 256, 256, 0, stream>>>(b_t1, c11b_w, c11b_b, b_x1);
  // 6. t = SimpleGate(c2a(x)) (3x3, g=32)
  conv3x3g32_gate_kernel<<<BB * 16 * PIX / 256, 256, 0, stream>>>(b_x, c2a_w, c2a_b, b_t);
  // 7. s2c = c2c(t)  (16 -> 32)
  conv1x1_wmma_kernel<2, 0><<<gemmGrid, 256, 0, stream>>>(b_t, c2c_w, c2c_b,
                                                          b_s2c, 16, nullptr, nullptr);
  // 8. c211o = c211(x) (64 -> 32)
  conv1x1_wmma_kernel<2, 0><<<gemmGrid, 256, 0, stream>>>(b_x, c211_w, c211_b,
                                                          b_c211, 64, nullptr, nullptr);
  // 9. uf0 = c1(x)
  conv1x1_wmma_kernel<4, 0><<<gemmGrid, 256, 0, stream>>>(b_x, c1_w, c1_b,
                                                          b_uf0, 64, nullptr, nullptr);
  // 10. uf = dw3x3(uf0)
  dw3x3_kernel<<<BB * CC * PIX / 256, 256, 0, stream>>>(b_uf0, c21_w, c21_b, b_uf);
  // 11. xk = (KBA(uf, att)*ga1 + uf) * x1 * sca   (att = s2c*attgamma + c211o)
  kba_kernel<<<BB * HH * (WW / 16), 256, 0, stream>>>(b_uf, b_s2c, b_c211, kba_w,
                                                      kba_b, attgamma, ga1, b_x1,
                                                      b_sca, b_xk);
  // 12. y = inp + c3(xk)*beta
  conv1x1_wmma_kernel<4, 1><<<gemmGrid, 256, 0, stream>>>(b_xk, c3_w, c3_b, b_y,
                                                          64, inp, beta);
  // 13. z = LN2(y)
  ln_kernel<<<BB * PIX / 256, 256, 0, stream>>>(b_y, norm2_w, norm2_b, b_z);
  // 14. zg = SimpleGate(c4(z))  (64 -> 128 -> 64)
  conv1x1_wmma_kernel<8, 2><<<gemmGrid, 256, 0, stream>>>(b_z, c4_w, c4_b, b_zg,
                                                          64, nullptr, nullptr);
  // 15. out = y + c5(zg)*gamma
  conv1x1_wmma_kernel<4, 1><<<gemmGrid, 256, 0, stream>>>(b_zg, c5_w, c5_b, out,
                                                          64, b_y, gamma);
}